// gpt2xcnn_75565654606376
// MI455X (gfx1250) — compile-verified
//
#include <hip/hip_runtime.h>
#include <hip/hip_bf16.h>
#include <math.h>

// ---------------- problem constants (from reference) ----------------
#define B_DIM     64
#define K_SIG     1044
#define T_DIM     500
#define KL        511
#define KR        533
#define VLEFT     1022
#define V_COLS    2089
#define FRLEN     64
#define FRSTEP    16
#define NF        28            // (500-64)/16 + 1
#define NBIN      33            // 64/2 + 1
#define FDIM      (3*NF*NBIN)   // 2772
#define HID       256
#define TWO_PI    6.283185307179586f

typedef __attribute__((ext_vector_type(2))) float v2f;
typedef __attribute__((ext_vector_type(8))) float v8f;

// ---------------- stage 1: motor[b,c,t] = sum_k tm[11+c, vidx(k)] * sigs[b,k,t]
// HBM-bound: reads sigs (134 MB) exactly once, coalesced over t.
__global__ void eeg_motor_kernel(const float* __restrict__ sigs,
                                 const float* __restrict__ tm,
                                 const int*   __restrict__ lidx,
                                 const int*   __restrict__ ridx,
                                 float*       __restrict__ motor) {
    int idx = blockIdx.x * blockDim.x + threadIdx.x;
    if (idx >= B_DIM * T_DIM) return;
    int b = idx / T_DIM;
    int t = idx - b * T_DIM;
    const float* tr0 = tm + (size_t)11 * V_COLS;
    const float* tr1 = tm + (size_t)12 * V_COLS;
    const float* tr2 = tm + (size_t)13 * V_COLS;
    const float* sb  = sigs + (size_t)b * K_SIG * T_DIM + t;
    float a0 = 0.f, a1 = 0.f, a2 = 0.f;
    for (int k = 0; k < K_SIG; ++k) {
        float v  = sb[(size_t)k * T_DIM];                 // coalesced over lanes (t)
        int  vi  = (k < KL) ? lidx[k] : (VLEFT + ridx[k - KL]); // wave-uniform, scalar-cached
        a0 = fmaf(tr0[vi], v, a0);
        a1 = fmaf(tr1[vi], v, a1);
        a2 = fmaf(tr2[vi], v, a2);
    }
    motor[((size_t)b * 3 + 0) * T_DIM + t] = a0;
    motor[((size_t)b * 3 + 1) * T_DIM + t] = a1;
    motor[((size_t)b * 3 + 2) * T_DIM + t] = a2;
}

// ---------------- stage 2: Hann window + 64-pt rDFT magnitude -> spec[b, c, f, bin]
__global__ void spec_kernel(const float* __restrict__ motor,
                            float*       __restrict__ spec) {
    int idx = blockIdx.x * blockDim.x + threadIdx.x;
    if (idx >= B_DIM * FDIM) return;
    int bin =  idx % NBIN;
    int f   = (idx / NBIN) % NF;
    int c   = (idx / (NBIN * NF)) % 3;
    int b   =  idx / (NBIN * NF * 3);
    const float* x = motor + ((size_t)b * 3 + c) * T_DIM + f * FRSTEP;
    float step = -TWO_PI * (float)bin * (1.0f / (float)FRLEN);
    float re = 0.f, im = 0.f;
    for (int n = 0; n < FRLEN; ++n) {
        float wn = 0.5f * (1.0f - __cosf(TWO_PI * (float)n * (1.0f / (float)FRLEN)));
        float v  = x[n] * wn;
        float s, cz;
        __sincosf(step * (float)n, &s, &cz);
        re = fmaf(v, cz, re);
        im = fmaf(v, s,  im);
    }
    spec[idx] = sqrtf(re * re + im * im);
}

// ---------------- stage 3: per-batch min/max over 2772 values
__global__ void minmax_kernel(const float* __restrict__ spec,
                              float* __restrict__ mn, float* __restrict__ mx) {
    __shared__ float smn[256];
    __shared__ float smx[256];
    int b   = blockIdx.x;
    int tid = threadIdx.x;
    float lo =  INFINITY, hi = -INFINITY;
    for (int i = tid; i < FDIM; i += 256) {
        float v = spec[(size_t)b * FDIM + i];
        lo = fminf(lo, v);
        hi = fmaxf(hi, v);
    }
    smn[tid] = lo; smx[tid] = hi;
    __syncthreads();
    for (int s = 128; s > 0; s >>= 1) {
        if (tid < s) {
            smn[tid] = fminf(smn[tid], smn[tid + s]);
            smx[tid] = fmaxf(smx[tid], smx[tid + s]);
        }
        __syncthreads();
    }
    if (tid == 0) { mn[b] = smn[0]; mx[b] = smx[0]; }
}

// ---------------- stage 4: h = relu(X @ W1 + b1) via V_WMMA_F32_16X16X4_F32
// X[b,k] = (spec[b,k]-mn[b]) / (mx[b]-mn[b]+1e-9), computed on the fly.
// One wave -> one 16x16 tile of h (M=64 -> 4 tiles, N=256 -> 16 tiles, 64 waves).
// ISA f32 layouts: A 16x4: lanes 0-15 carry K={k,k+1}, lanes 16-31 K={k+2,k+3},
// M = lane&15.  B 4x16 / C 16x16: row-striped across lanes with half-split.
__global__ void mlp1_wmma_kernel(const float* __restrict__ spec,
                                 const float* __restrict__ mn,
                                 const float* __restrict__ mx,
                                 const float* __restrict__ W1,
                                 const float* __restrict__ b1,
                                 float*       __restrict__ h) {
    int lane = threadIdx.x & 31;
    int wave = threadIdx.x >> 5;
    int tile = blockIdx.x * (blockDim.x >> 5) + wave;   // 0..63, exact
    int mt   = tile >> 4;                               // 0..3  (M tile)
    int nt   = tile & 15;                               // 0..15 (N tile)
    int half = lane >> 4;                               // 0 or 1
    int l15  = lane & 15;
    int n    = nt * 16 + l15;                           // output column
    int mrow = mt * 16 + l15;                           // A row for this lane

    float mnv = mn[mrow];
    float inv = 1.0f / (mx[mrow] - mnv + 1e-9f);
    const float* xrow = spec + (size_t)mrow * FDIM;

    v8f acc = {};
    for (int k0 = 0; k0 < FDIM; k0 += 4) {              // 693 clean steps
        int ka = k0 + half * 2;
        v2f a, bm;
        a.x  = (xrow[ka]     - mnv) * inv;
        a.y  = (xrow[ka + 1] - mnv) * inv;
        bm.x = W1[(size_t)(ka)     * HID + n];
        bm.y = W1[(size_t)(ka + 1) * HID + n];
        acc = __builtin_amdgcn_wmma_f32_16x16x4_f32(
                  /*neg_a=*/false, a, /*neg_b=*/false, bm,
                  /*c_mod=*/(short)0, acc,
                  /*reuse_a=*/false, /*reuse_b=*/false);
    }

    float bias = b1[n];
    #pragma unroll
    for (int r = 0; r < 8; ++r) {
        int m = mt * 16 + r + half * 8;                 // C layout: VGPR r -> rows r / r+8
        h[(size_t)m * HID + n] = fmaxf(acc[r] + bias, 0.0f);
    }
}

// ---------------- stage 5: y = sigmoid(h @ W2 + b2); BCE loss; out = [loss, y]
__global__ void head_kernel(const float* __restrict__ h,
                            const float* __restrict__ W2,
                            const float* __restrict__ b2,
                            const int*   __restrict__ labels,
                            float*       __restrict__ out) {
    __shared__ float red[64];
    int b = threadIdx.x;                                // 0..63
    float s = 0.f;
    for (int k = 0; k < HID; ++k)
        s = fmaf(h[(size_t)b * HID + k], W2[k], s);
    float y = 1.0f / (1.0f + expf(-(s + b2[0])));
    out[1 + b] = y;
    float lab  = (float)labels[b];
    const float eps = 1e-7f;
    float yc   = fminf(fmaxf(y, eps), 1.0f - eps);
    red[b] = -(lab * logf(yc) + (1.0f - lab) * log1pf(-yc));
    __syncthreads();
    for (int st = 32; st > 0; st >>= 1) {
        if (b < st) red[b] += red[b + st];
        __syncthreads();
    }
    if (b == 0) out[0] = red[0] * (1.0f / (float)B_DIM);
}

extern "C" void kernel_launch(void* const* d_in, const int* in_sizes, int n_in,
                              void* d_out, int out_size, void* d_ws, size_t ws_size,
                              hipStream_t stream) {
    (void)in_sizes; (void)n_in; (void)out_size; (void)ws_size;
    const float* sigs  = (const float*)d_in[0];
    const float* tm    = (const float*)d_in[1];
    const float* W1    = (const float*)d_in[2];
    const float* b1    = (const float*)d_in[3];
    const float* W2    = (const float*)d_in[4];
    const float* b2    = (const float*)d_in[5];
    const int*   lidx  = (const int*)  d_in[6];
    const int*   ridx  = (const int*)  d_in[7];
    const int*   lab   = (const int*)  d_in[8];
    float* out = (float*)d_out;

    // workspace layout (floats): motor | spec | mn | mx | h  (~1.16 MB total)
    float* ws    = (float*)d_ws;
    float* motor = ws;                               // 64*3*500   = 96000
    float* spec  = motor + (size_t)B_DIM * 3 * T_DIM;// 64*2772    = 177408
    float* mn    = spec  + (size_t)B_DIM * FDIM;     // 64
    float* mx    = mn + B_DIM;                       // 64
    float* h     = mx + B_DIM;                       // 64*256     = 16384

    {   // stage 1: motor projection (HBM-bound, reads sigs once)
        int n = B_DIM * T_DIM;
        eeg_motor_kernel<<<(n + 255) / 256, 256, 0, stream>>>(sigs, tm, lidx, ridx, motor);
    }
    {   // stage 2: windowed rDFT magnitudes
        int n = B_DIM * FDIM;
        spec_kernel<<<(n + 255) / 256, 256, 0, stream>>>(motor, spec);
    }
    // stage 3: per-batch min/max
    minmax_kernel<<<B_DIM, 256, 0, stream>>>(spec, mn, mx);
    // stage 4: WMMA GEMM 64x2772 @ 2772x256 + bias + ReLU (8 blocks x 8 waves = 64 tiles)
    mlp1_wmma_kernel<<<8, 256, 0, stream>>>(spec, mn, mx, W1, b1, h);
    // stage 5: head + sigmoid + BCE -> out[0]=loss, out[1..64]=y
    head_kernel<<<1, 64, 0, stream>>>(h, W2, b2, lab, out);
}